// GATNetwork_46626164966102
// MI455X (gfx1250) — compile-verified
//
#include <hip/hip_runtime.h>
#include <hip/hip_bf16.h>

// ---------------------------------------------------------------------------
// GAT forward for MI455X (gfx1250, wave32, WMMA + TDM).
// All matmuls use V_WMMA_F32_16X16X4_F32 (fp32 A/B/C -> numerically matches
// the fp32 reference). g_l is never materialized (score_l = hp @ (W_l @ a_l)).
// g_r is recomputed per (s,p,h) from the LDS-resident h-slice, so HBM traffic
// is just: read h once, write out + a once (~235 MB -> ~10us at 23.3 TB/s).
// The h-slice (128x256 fp32, row stride S*P*F) is brought into LDS by the
// Tensor Data Mover with hardware padding to the 260-float LDS row stride.
// ---------------------------------------------------------------------------

typedef float v2f __attribute__((ext_vector_type(2)));
typedef float v8f __attribute__((ext_vector_type(8)));
typedef unsigned int v4u __attribute__((ext_vector_type(4)));
typedef int v4i __attribute__((ext_vector_type(4)));
typedef int v8i __attribute__((ext_vector_type(8)));

#define NN_   128   // nodes
#define SS_   32
#define PP_   16
#define FF_   256
#define HH_   4
#define SLOPE 0.2f

// LDS layout (floats). Strides padded for 16B alignment + bank spread.
#define HS_STRIDE 260                         // 128 x 256 h-slice rows
#define GT_STRIDE 130                         // transposed Gh (128 x 128)
#define AL_STRIDE 130                         // per-wave softmax tile 16 x 128
#define HS_OFF 0
#define GT_OFF (128 * HS_STRIDE)              // 33280
#define AL_OFF (GT_OFF + 128 * GT_STRIDE)     // 49920
#define SL_OFF (AL_OFF + 8 * 16 * AL_STRIDE)  // 66560  scoreL [h][i]
#define SR_OFF (SL_OFF + 512)                 // 67072  scoreR [h][i]
#define U_OFF  (SR_OFF + 512)                 // 67584  u_l (1024) + u_r (1024)
#define SMEM_FLOATS (U_OFF + 2048)            // 69632
#define SMEM_BYTES  (SMEM_FLOATS * 4)         // 278528 B  (< 320 KB WGP LDS)

// --------------------------- kernel A: u = W @ a ---------------------------
__global__ void gat_u_kernel(const float* __restrict__ Wl,
                             const float* __restrict__ Wr,
                             const float* __restrict__ attn,
                             float* __restrict__ u) {
  int c = blockIdx.x * blockDim.x + threadIdx.x;
  if (c >= FF_) return;
#pragma unroll
  for (int h = 0; h < HH_; ++h) {
    float sl = 0.f, sr = 0.f;
    for (int f = 0; f < 128; ++f) {
      sl += Wl[c * 512 + h * 128 + f] * attn[f];
      sr += Wr[c * 512 + h * 128 + f] * attn[128 + f];
    }
    u[c * 4 + h]        = sl;   // u_l[c][h]
    u[1024 + c * 4 + h] = sr;   // u_r[c][h]
  }
}

// ---- TDM: DMA the (128 x 256 fp32) h-slice into LDS with 4-DWORD row pad ----
// D# per CDNA5 ISA ch.8: 2D tile, data_size=4B, tensor/tile dims 256 x 128,
// dim0 stride = S*P*F = 131072 elements; pad_interval=256 DW, pad_amount=4 DW
// -> LDS row stride 260 floats (== HS_STRIDE).
__device__ __forceinline__ void tdm_load_h_slice(const float* gptr, float* lds_ptr) {
  unsigned long long ga = (unsigned long long)(uintptr_t)gptr;
  unsigned int lds = (unsigned int)(uintptr_t)lds_ptr;  // low 32b = LDS offset

  v4u g0;
  g0.x = 0x1u;                                              // count=1, user D#
  g0.y = lds;                                               // lds_addr (bytes)
  g0.z = (unsigned int)ga;                                  // global_addr[31:0]
  g0.w = ((unsigned int)(ga >> 32) & 0x01FFFFFFu)           // global_addr[56:32]
         | (2u << 30);                                      // type = 2 (image)

  v8i g1;
  g1[0] = (int)((2u << 16)        // data_size = 4 bytes
              | (1u << 20)        // pad_enable
              | (7u << 22)        // pad_interval code 7 = 256 DWORDs
              | (3u << 25));      // pad_amount  code 3 = 4 DWORDs
  g1[1] = (int)(256u << 16);      // tensor_dim0 = 256 (bits 79:48 low half)
  g1[2] = (int)(128u << 16);      // tensor_dim0 hi=0, tensor_dim1 = 128
  g1[3] = (int)(256u << 16);      // tensor_dim1 hi=0, tile_dim0 = 256
  g1[4] = 128;                    // tile_dim1 = 128, tile_dim2 = 0
  g1[5] = 131072;                 // tensor_dim0_stride[31:0] (elements)
  g1[6] = 0;                      // stride hi, tensor_dim1_stride lo
  g1[7] = 0;

  v4i z4 = {0, 0, 0, 0};          // groups 2/3: unused for 2D tiles
#if __clang_major__ >= 23
  v8i z8 = {0, 0, 0, 0, 0, 0, 0, 0};
  __builtin_amdgcn_tensor_load_to_lds(g0, g1, z4, z4, z8, 0);
#else
  __builtin_amdgcn_tensor_load_to_lds(g0, g1, z4, z4, 0);
#endif
}

// --------------------------- kernel B: fused GAT ---------------------------
__global__ __launch_bounds__(256, 1)
void gat_main_kernel(const float* __restrict__ hten,
                     const float* __restrict__ Wr,
                     const float* __restrict__ uG,
                     float* __restrict__ out,
                     float* __restrict__ a_out) {
  extern __shared__ float sm[];
  float* sm_h  = sm + HS_OFF;
  float* sm_gt = sm + GT_OFF;
  float* sm_a  = sm + AL_OFF;
  float* sm_sl = sm + SL_OFF;
  float* sm_sr = sm + SR_OFF;
  float* sm_u  = sm + U_OFF;

  const int tid  = threadIdx.x;
  const int lane = tid & 31;
  const int wv   = tid >> 5;        // 8 waves
  const int m    = lane & 15;       // fragment row/col index
  const int hi   = lane >> 4;       // half-wave select
  const int sp   = blockIdx.x;      // s*16 + p

  const v8f vzero = {0.f, 0.f, 0.f, 0.f, 0.f, 0.f, 0.f, 0.f};

  // ---- TDM DMA: h slice hS[n][f] = h[n, s, p, f] -> padded LDS rows ----
  const float* hb = hten + (size_t)sp * FF_;
  if (wv == 0) {
    tdm_load_h_slice(hb, sm_h);
  }

  // ---- load u_l / u_r (2048 floats) while the DMA runs ----
#pragma unroll
  for (int k = 0; k < 8; ++k) sm_u[tid * 8 + k] = uG[tid * 8 + k];

  if (wv == 0) {
    __builtin_amdgcn_s_wait_tensorcnt(0);   // issuing wave sees DMA done
  }
  __syncthreads();                          // everyone sees hS + u

  // ---- scores: score_{l,r}[h][i] = sum_c hS[i][c] * u_{l,r}[c][h] ----
  {
    int i    = tid >> 1;
    int base = (tid & 1) * 128;
    float accl[HH_] = {0.f, 0.f, 0.f, 0.f};
    float accr[HH_] = {0.f, 0.f, 0.f, 0.f};
    for (int c = 0; c < 128; ++c) {
      float hv = sm_h[i * HS_STRIDE + base + c];
      const float* ul = &sm_u[(base + c) * 4];
      const float* ur = &sm_u[1024 + (base + c) * 4];
#pragma unroll
      for (int h = 0; h < HH_; ++h) {
        accl[h] += hv * ul[h];
        accr[h] += hv * ur[h];
      }
    }
#pragma unroll
    for (int h = 0; h < HH_; ++h) {
      accl[h] += __shfl_xor(accl[h], 1);
      accr[h] += __shfl_xor(accr[h], 1);
    }
    if ((tid & 1) == 0) {
#pragma unroll
      for (int h = 0; h < HH_; ++h) {
        sm_sl[h * 128 + i] = accl[h];
        sm_sr[h * 128 + i] = accr[h];
      }
    }
  }
  __syncthreads();

  // ---- per-head loop: Gh = hS @ Wr[:,h*128:], softmax, out += a @ Gh ----
  v8f acc_out[8];
#pragma unroll
  for (int ft = 0; ft < 8; ++ft) acc_out[ft] = vzero;

  const int j0 = wv * 16;   // this wave's 16-row j-tile (== its i-tile)

  for (int hh = 0; hh < HH_; ++hh) {
    // ---------- GEMM1: Gh[j][f] = sum_c hS[j][c] * Wr[c][hh*128+f] ----------
    const float* Wb = Wr + hh * 128;
#pragma unroll 2
    for (int ft = 0; ft < 8; ++ft) {
      v8f acc = vzero;
      int col = ft * 16 + m;            // N index (f within head)
#pragma unroll 4
      for (int kb = 0; kb < 64; ++kb) {
        int k0 = kb * 4 + 2 * hi;
        v2f av = *(const v2f*)&sm_h[(j0 + m) * HS_STRIDE + k0];
        v2f bv;
        bv.x = Wb[k0 * 512 + col];
        bv.y = Wb[(k0 + 1) * 512 + col];
        acc = __builtin_amdgcn_wmma_f32_16x16x4_f32(
            false, av, false, bv, (short)0, acc, false, false);
      }
      // store transposed into LDS: GhT[f][j]; lane holds N=col fixed,
      // M = j0 + v + 8*hi for vgpr v -> consecutive columns.
      int gtb = (ft * 16 + m) * GT_STRIDE + j0 + 8 * hi;
#pragma unroll
      for (int v2 = 0; v2 < 4; ++v2) {
        v2f t;
        t.x = acc[2 * v2];
        t.y = acc[2 * v2 + 1];
        *(v2f*)&sm_gt[gtb + 2 * v2] = t;
      }
    }
    __syncthreads();   // all waves' j-stripes of GhT visible

    // ---------- softmax rows for this wave's i-tile ----------
    for (int r = 0; r < 16; ++r) {
      int i = j0 + r;
      float sR = sm_sr[hh * 128 + i];
      float e[4];
      float mx = -3.4e38f;
#pragma unroll
      for (int q = 0; q < 4; ++q) {
        int j = q * 32 + lane;
        float v = sR + sm_sl[hh * 128 + j];
        v = (v >= 0.f) ? v : SLOPE * v;
        if (j == i) v = -__builtin_inff();
        e[q] = v;
        mx = fmaxf(mx, v);
      }
#pragma unroll
      for (int off = 16; off >= 1; off >>= 1) mx = fmaxf(mx, __shfl_xor(mx, off));
      float s = 0.f;
#pragma unroll
      for (int q = 0; q < 4; ++q) {
        e[q] = __expf(e[q] - mx);
        s += e[q];
      }
#pragma unroll
      for (int off = 16; off >= 1; off >>= 1) s += __shfl_xor(s, off);
      float inv = 1.f / s;
      size_t ab = (size_t)sp * 65536 + (size_t)i * 512 + hh;
#pragma unroll
      for (int q = 0; q < 4; ++q) {
        float pv = e[q] * inv;
        int j = q * 32 + lane;
        sm_a[wv * (16 * AL_STRIDE) + r * AL_STRIDE + j] = pv;
        a_out[ab + (size_t)j * 4] = pv;    // a[s,p,i,j,h]
      }
    }

    // ---------- out-GEMM: acc_out[i][f] += sum_j a[i][j] * Gh[j][f] ----------
#pragma unroll 2
    for (int ft = 0; ft < 8; ++ft) {
      v8f acc = acc_out[ft];
#pragma unroll 4
      for (int kb = 0; kb < 32; ++kb) {
        int k0 = kb * 4 + 2 * hi;
        v2f av = *(const v2f*)&sm_a[wv * (16 * AL_STRIDE) + m * AL_STRIDE + k0];
        v2f bv = *(const v2f*)&sm_gt[(ft * 16 + m) * GT_STRIDE + k0];
        acc = __builtin_amdgcn_wmma_f32_16x16x4_f32(
            false, av, false, bv, (short)0, acc, false, false);
      }
      acc_out[ft] = acc;
    }
    __syncthreads();   // before GhT is overwritten next head
  }

  // ---- store out[n,s,p,f] = mean over heads ----
#pragma unroll
  for (int ft = 0; ft < 8; ++ft) {
    int f = ft * 16 + m;
#pragma unroll
    for (int v = 0; v < 8; ++v) {
      int i = j0 + v + 8 * hi;
      out[(size_t)i * 65536 + (size_t)sp * 128 + f] = acc_out[ft][v] * 0.25f;
    }
  }
}

// ------------------------------- launcher ----------------------------------
extern "C" void kernel_launch(void* const* d_in, const int* in_sizes, int n_in,
                              void* d_out, int out_size, void* d_ws, size_t ws_size,
                              hipStream_t stream) {
  (void)in_sizes; (void)n_in; (void)out_size; (void)ws_size;
  const float* h    = (const float*)d_in[0];
  const float* Wl   = (const float*)d_in[1];
  const float* Wr   = (const float*)d_in[2];
  const float* attn = (const float*)d_in[3];

  float* out   = (float*)d_out;
  float* a_out = out + (size_t)NN_ * SS_ * PP_ * 128;   // second output (a)
  float* u     = (float*)d_ws;                           // 8 KB scratch

  gat_u_kernel<<<1, 256, 0, stream>>>(Wl, Wr, attn, u);

  hipFuncSetAttribute(reinterpret_cast<const void*>(gat_main_kernel),
                      hipFuncAttributeMaxDynamicSharedMemorySize, SMEM_BYTES);
  gat_main_kernel<<<SS_ * PP_, 256, SMEM_BYTES, stream>>>(h, Wr, u, out, a_out);
}